// SocialLSTMNet_32469952758509
// MI455X (gfx1250) — compile-verified
//
#include <hip/hip_runtime.h>
#include <cstdint>
#include <cstddef>

#define HDIM 64
#define TENC 8
#define PREDN 12
#define KTOT 192          // 128 (emb|social) + 64 (h)
#define KSTEP 48          // KTOT / 4  (K=4 per V_WMMA_F32_16X16X4_F32)
#define GATES 256

#define POOL_BLOCKS 512
#define POOL_THREADS 1024   // 32 waves/block -> 16384 waves total
#define CNT_BLOCKS 1024
#define CNT_THREADS 256

typedef __attribute__((ext_vector_type(2))) float v2f;
typedef __attribute__((ext_vector_type(8))) float v8f;

// ---------------------------------------------------------------------------
// ws layout:  [0,64)           : 16 x uint32 cell counts (64B pad)
//             [64, 64+512*64*4): per-block pooled partial sums (float)
// ---------------------------------------------------------------------------

__global__ void zero_ws_kernel(unsigned* __restrict__ cnt) {
  if (threadIdx.x < 16) cnt[threadIdx.x] = 0u;
}

__device__ __forceinline__ bool cell_of(float dx, float dy, int& cell) {
  if (fmaxf(fabsf(dx), fabsf(dy)) > 4.0f) return false;
  // (d/NB + 1)/2 * G == d*0.5 + 2  for NB=4, G=4
  int gx = (int)floorf(dx * 0.5f + 2.0f); gx = min(max(gx, 0), 3);
  int gy = (int)floorf(dy * 0.5f + 2.0f); gy = min(max(gy, 0), 3);
  cell = gy * 4 + gx;
  return true;
}

// ---------------- pass 1: neighbor counts per grid cell (integer, exact) ----
__global__ void count_kernel(const float* __restrict__ pos,
                             const int* __restrict__ idxp,
                             unsigned* __restrict__ cnt, int N) {
  __shared__ unsigned lc[16];
  if (threadIdx.x < 16) lc[threadIdx.x] = 0u;
  __syncthreads();
  const int idx = *idxp;
  const float px0 = pos[2 * idx], py0 = pos[2 * idx + 1];
  const float2* p2 = (const float2*)pos;
  const int stride = gridDim.x * blockDim.x;
  for (int a = blockIdx.x * blockDim.x + threadIdx.x; a < N; a += stride) {
    float2 p = p2[a];
    int cell;
    if (a != idx && cell_of(p.x - px0, p.y - py0, cell))
      atomicAdd(&lc[cell], 1u);
  }
  __syncthreads();
  if (threadIdx.x < 16) atomicAdd(&cnt[threadIdx.x], lc[threadIdx.x]);
}

// ---------------- pass 2: stream all_h, accumulate sum(h / count[cell]) -----
// wave-per-agent, 4 agents batched per iteration so up to 4 rows (plus 4
// prefetched rows) are in flight per wave; lane l owns h[2l], h[2l+1].
__global__ void __launch_bounds__(POOL_THREADS)
pool_kernel(const float* __restrict__ all_h,
            const float* __restrict__ pos,
            const int* __restrict__ idxp,
            const unsigned* __restrict__ cnt,
            float* __restrict__ partials, int N) {
  const int wib = threadIdx.x >> 5;
  const int l   = threadIdx.x & 31;
  const int nwb = blockDim.x >> 5;                 // 32 waves / block
  const int gw  = blockIdx.x * nwb + wib;
  const int tw  = gridDim.x * nwb;                 // total waves
  const int idx = *idxp;
  const float px0 = pos[2 * idx], py0 = pos[2 * idx + 1];
  const float2* p2 = (const float2*)pos;

  float ax = 0.f, ay = 0.f;
  int a = gw;
  for (; a + 3 * tw < N; a += 4 * tw) {
    const int a0 = a, a1 = a + tw, a2 = a + 2 * tw, a3 = a + 3 * tw;
    // prefetch next batch's row slices (speculative, no counter cost)
    {
      int an = a + 4 * tw;
      if (an            < N) __builtin_prefetch(all_h + (size_t)an            * HDIM + 2 * l, 0, 0);
      if (an + tw       < N) __builtin_prefetch(all_h + (size_t)(an + tw)     * HDIM + 2 * l, 0, 0);
      if (an + 2 * tw   < N) __builtin_prefetch(all_h + (size_t)(an + 2 * tw) * HDIM + 2 * l, 0, 0);
      if (an + 3 * tw   < N) __builtin_prefetch(all_h + (size_t)(an + 3 * tw) * HDIM + 2 * l, 0, 0);
    }
    float2 q0 = p2[a0], q1 = p2[a1], q2 = p2[a2], q3 = p2[a3];
    int c0 = 0, c1 = 0, c2 = 0, c3 = 0;
    const bool m0 = (a0 != idx) && cell_of(q0.x - px0, q0.y - py0, c0);
    const bool m1 = (a1 != idx) && cell_of(q1.x - px0, q1.y - py0, c1);
    const bool m2 = (a2 != idx) && cell_of(q2.x - px0, q2.y - py0, c2);
    const bool m3 = (a3 != idx) && cell_of(q3.x - px0, q3.y - py0, c3);
    float2 h0 = {0.f, 0.f}, h1 = {0.f, 0.f}, h2 = {0.f, 0.f}, h3 = {0.f, 0.f};
    float s0 = 0.f, s1 = 0.f, s2 = 0.f, s3 = 0.f;
    // uniform branches; loads issue back-to-back, waits deferred to the FMAs
    if (m0) { s0 = 1.0f / fmaxf((float)cnt[c0], 1.0f);
              h0 = ((const float2*)(all_h + (size_t)a0 * HDIM))[l]; }
    if (m1) { s1 = 1.0f / fmaxf((float)cnt[c1], 1.0f);
              h1 = ((const float2*)(all_h + (size_t)a1 * HDIM))[l]; }
    if (m2) { s2 = 1.0f / fmaxf((float)cnt[c2], 1.0f);
              h2 = ((const float2*)(all_h + (size_t)a2 * HDIM))[l]; }
    if (m3) { s3 = 1.0f / fmaxf((float)cnt[c3], 1.0f);
              h3 = ((const float2*)(all_h + (size_t)a3 * HDIM))[l]; }
    ax = fmaf(h0.x, s0, ax); ay = fmaf(h0.y, s0, ay);
    ax = fmaf(h1.x, s1, ax); ay = fmaf(h1.y, s1, ay);
    ax = fmaf(h2.x, s2, ax); ay = fmaf(h2.y, s2, ay);
    ax = fmaf(h3.x, s3, ax); ay = fmaf(h3.y, s3, ay);
  }
  for (; a < N; a += tw) {                         // tail
    float2 p = p2[a];
    int cell;
    if (a != idx && cell_of(p.x - px0, p.y - py0, cell)) {
      float s = 1.0f / fmaxf((float)cnt[cell], 1.0f);
      float2 hv = ((const float2*)(all_h + (size_t)a * HDIM))[l];
      ax = fmaf(hv.x, s, ax);
      ay = fmaf(hv.y, s, ay);
    }
  }

  __shared__ float red[(POOL_THREADS / 32) * HDIM];
  red[wib * HDIM + 2 * l]     = ax;
  red[wib * HDIM + 2 * l + 1] = ay;
  __syncthreads();
  if (threadIdx.x < HDIM) {                        // fixed-order, deterministic
    float s = 0.f;
    #pragma unroll
    for (int wq = 0; wq < POOL_THREADS / 32; ++wq)
      s += red[wq * HDIM + threadIdx.x];
    partials[blockIdx.x * HDIM + threadIdx.x] = s;
  }
}

// ---------------- pass 3: social matvec + serial LSTM via f32 WMMA ----------
__device__ __forceinline__ float sigm(float x) { return 1.0f / (1.0f + expf(-x)); }

__global__ void __launch_bounds__(512)
head_kernel(const float* __restrict__ history,
            const float* __restrict__ W_in, const float* __restrict__ b_in,
            const float* __restrict__ W_ih, const float* __restrict__ b_ih,
            const float* __restrict__ W_hh, const float* __restrict__ b_hh,
            const float* __restrict__ W_sp, const float* __restrict__ b_sp,
            const float* __restrict__ W_out, const float* __restrict__ b_out,
            const float* __restrict__ partials,
            float* __restrict__ out) {
  __shared__ float vsh[KTOT];    // [emb(64) | social(64) | h(64)]
  __shared__ float csh[HDIM];
  __shared__ float gsh[GATES];
  __shared__ float pl[HDIM];
  __shared__ float dlt[2];
  __shared__ float lp[2];

  const int tid  = threadIdx.x;
  const int w    = tid >> 5;     // wave -> owns gate rows [16w, 16w+16)
  const int l    = tid & 31;
  const int mrow = l & 15;
  const bool hi  = (l >= 16);

  // --- per-lane A-operand preload: Wcat = [W_ih(256x128) | W_hh(256x64)] ---
  // 16x4 f32 A layout: lanes 0-15 V0=K0,V1=K1; lanes 16-31 V0=K2,V1=K3.
  float aw[2 * KSTEP];
  {
    const int r = w * 16 + mrow;
    const int koff = hi ? 2 : 0;
    #pragma unroll
    for (int k = 0; k < KSTEP; ++k) {
      if (k < 32) {
        float2 t = *(const float2*)(W_ih + r * 128 + 4 * k + koff);
        aw[2 * k] = t.x; aw[2 * k + 1] = t.y;
      } else {
        float2 t = *(const float2*)(W_hh + r * 64 + 4 * (k - 32) + koff);
        aw[2 * k] = t.x; aw[2 * k + 1] = t.y;
      }
    }
  }
  // fused bias in C layout: VGPR m holds M=m (lanes 0-15) / M=8+m (16-31)
  float bs[8];
  #pragma unroll
  for (int m = 0; m < 8; ++m) {
    int r = w * 16 + m + (hi ? 8 : 0);
    bs[m] = b_ih[r] + b_hh[r];
  }

  // --- deterministic reduction of pooled partials, then social matvec ---
  if (tid < HDIM) {
    float s = 0.f;
    for (int b = 0; b < POOL_BLOCKS; ++b) s += partials[b * HDIM + tid];
    pl[tid] = s;
  }
  __syncthreads();
  if (tid < HDIM) {
    float s = b_sp[tid];
    for (int j = 0; j < HDIM; ++j) s = fmaf(W_sp[tid * HDIM + j], pl[j], s);
    vsh[64 + tid]  = s;     // social (fixed all steps)
    vsh[128 + tid] = 0.f;   // h0
    csh[tid]       = 0.f;   // c0
  }
  if (tid == 0) { lp[0] = history[28]; lp[1] = history[29]; }
  __syncthreads();

  // gates[16w..16w+16) = Wcat_tile @ vsh + bias, via 48 chained f32 WMMAs.
  // B (4x16): row K striped across lane halves per VGPR -> all 16 columns
  // replicated with the same input chunk, so every D column equals the GEMV.
  auto lstm_wmma = [&]() {
    v8f acc;
    #pragma unroll
    for (int m = 0; m < 8; ++m) acc[m] = bs[m];
    const int bo = hi ? 1 : 0;
    #pragma unroll
    for (int k = 0; k < KSTEP; ++k) {
      v2f A; A[0] = aw[2 * k]; A[1] = aw[2 * k + 1];
      v2f B; B[0] = vsh[4 * k + bo]; B[1] = vsh[4 * k + bo + 2];
      acc = __builtin_amdgcn_wmma_f32_16x16x4_f32(
          false, A, false, B, (short)0, acc, false, false);
    }
    if (l == 0) {
      #pragma unroll
      for (int m = 0; m < 8; ++m) gsh[w * 16 + m] = acc[m];
    } else if (l == 16) {
      #pragma unroll
      for (int m = 0; m < 8; ++m) gsh[w * 16 + 8 + m] = acc[m];
    }
  };

  auto lstm_update = [&]() {   // i,f,g,o split; c,h update
    if (tid < HDIM) {
      float ig = sigm(gsh[tid]);
      float fg = sigm(gsh[64 + tid]);
      float gg = tanhf(gsh[128 + tid]);
      float og = sigm(gsh[192 + tid]);
      float cn = fmaf(fg, csh[tid], ig * gg);
      csh[tid] = cn;
      vsh[128 + tid] = og * tanhf(cn);
    }
  };

  // --- encoder: 8 steps over history ---
  for (int t = 0; t < TENC; ++t) {
    if (tid < HDIM) {
      float e = b_in[tid];
      #pragma unroll
      for (int q = 0; q < 4; ++q)
        e = fmaf(W_in[tid * 4 + q], history[t * 4 + q], e);
      vsh[tid] = e;
    }
    __syncthreads();
    lstm_wmma();
    __syncthreads();
    lstm_update();
    __syncthreads();
  }

  // --- decoder: 12 prediction steps ---
  for (int s = 0; s < PREDN; ++s) {
    if (tid < 2) {
      float d = b_out[tid];
      for (int j = 0; j < HDIM; ++j)
        d = fmaf(W_out[tid * HDIM + j], vsh[128 + j], d);
      dlt[tid] = d;
      out[2 * s + tid] = d;                 // preds[s] = delta
    }
    __syncthreads();
    if (tid < HDIM) {
      float x0 = lp[0] + dlt[0];
      float x1 = lp[1] + dlt[1];
      float x2 = dlt[0] * 2.0f;             // delta / 0.5
      float x3 = dlt[1] * 2.0f;
      float e = b_in[tid];
      e = fmaf(W_in[tid * 4 + 0], x0, e);
      e = fmaf(W_in[tid * 4 + 1], x1, e);
      e = fmaf(W_in[tid * 4 + 2], x2, e);
      e = fmaf(W_in[tid * 4 + 3], x3, e);
      vsh[tid] = e;
    }
    __syncthreads();
    if (tid < 2) lp[tid] += dlt[tid];       // carry update (read next iter)
    lstm_wmma();
    __syncthreads();
    lstm_update();
    __syncthreads();
  }
}

// ---------------------------------------------------------------------------
extern "C" void kernel_launch(void* const* d_in, const int* in_sizes, int n_in,
                              void* d_out, int out_size, void* d_ws, size_t ws_size,
                              hipStream_t stream) {
  const float* history = (const float*)d_in[0];
  const float* all_h   = (const float*)d_in[1];
  const float* pos     = (const float*)d_in[2];
  const float* W_in    = (const float*)d_in[3];
  const float* b_in    = (const float*)d_in[4];
  const float* W_ih    = (const float*)d_in[5];
  const float* b_ih    = (const float*)d_in[6];
  const float* W_hh    = (const float*)d_in[7];
  const float* b_hh    = (const float*)d_in[8];
  const float* W_sp    = (const float*)d_in[9];
  const float* b_sp    = (const float*)d_in[10];
  const float* W_out   = (const float*)d_in[11];
  const float* b_out   = (const float*)d_in[12];
  const int*   idxp    = (const int*)d_in[13];
  float* out = (float*)d_out;
  const int N = in_sizes[2] / 2;

  unsigned* cnt   = (unsigned*)d_ws;
  float* partials = (float*)((char*)d_ws + 64);

  zero_ws_kernel<<<1, 32, 0, stream>>>(cnt);
  count_kernel<<<CNT_BLOCKS, CNT_THREADS, 0, stream>>>(pos, idxp, cnt, N);
  pool_kernel<<<POOL_BLOCKS, POOL_THREADS, 0, stream>>>(all_h, pos, idxp, cnt,
                                                        partials, N);
  head_kernel<<<1, 512, 0, stream>>>(history, W_in, b_in, W_ih, b_ih, W_hh,
                                     b_hh, W_sp, b_sp, W_out, b_out, partials,
                                     out);
}